// LogicLayer_58763742544750
// MI455X (gfx1250) — compile-verified
//
#include <hip/hip_runtime.h>
#include <hip/hip_bf16.h>
#include <math.h>

typedef __attribute__((ext_vector_type(2))) float v2f;
typedef __attribute__((ext_vector_type(8))) float v8f;

// 16x4 gate-coefficient matrix (f32), flat [k*4 + c], same as Python GATE_COEFFS.
__constant__ float GATE_COEFFS_C[64] = {
    0, 0, 0, 0,   0, 0, 0, 1,   0, 1, 0, -1,  0, 1, 0, 0,
    0, 0, 1, -1,  0, 0, 1, 0,   0, 1, 1, -2,  0, 1, 1, -1,
    1, -1, -1, 1, 1, -1, -1, 2, 1, 0, -1, 0,  1, 0, -1, 1,
    1, -1, 0, 0,  1, -1, 0, 1,  1, 0, 0, -1,  1, 0, 0, 0};

// ---------------------------------------------------------------------------
// Phase 1: coeffs[j][0:4] = softmax(weights[j][0:16]) @ GATE_COEFFS
// One wave per tile of 16 rows; K=16 done as four accumulating
// V_WMMA_F32_16X16X4_F32 steps (exact f32, same arithmetic as the reference).
//
// ISA layouts (CDNA5 ISA):
//   A (16x4 f32):  lane L -> row M = L%16 ; VGPR0 K = 2*(L/16), VGPR1 K = +1.
//   B (4x16 f32):  VGPR0: lanes 0-15 K=0 / lanes 16-31 K=2 (N = L%16);
//                  VGPR1: K=1 / K=3.
//   D (16x16 f32): lane L, vgpr v -> (M = v + 8*(L/16), N = L%16).
// Only D columns N=0..3 are read back; B columns N>=4 are therefore
// don't-care, so we load GATE_COEFFS[ka][n&3] unconditionally (no exec mask).
// ---------------------------------------------------------------------------
__global__ __launch_bounds__(32) void coeffs_wmma_kernel(
    const float* __restrict__ weights,  // [OUT_DIM][16]
    float* __restrict__ coeffs)         // [OUT_DIM][4]
{
    const int lane = threadIdx.x;       // 0..31
    const int n    = lane & 15;         // A row index / D column index
    const int half = lane >> 4;         // 0 or 1
    const int tile = blockIdx.x;
    const int row  = tile * 16 + n;

    // Per-lane f32 softmax of row (lane%16); both half-waves compute the same
    // row redundantly — trivial cost, keeps EXEC all-ones for WMMA.
    const float* wr = weights + (size_t)row * 16;
    float w[16];
    float m = -INFINITY;
#pragma unroll
    for (int k = 0; k < 16; ++k) { w[k] = wr[k]; m = fmaxf(m, w[k]); }
    float s = 0.f;
#pragma unroll
    for (int k = 0; k < 16; ++k) { w[k] = expf(w[k] - m); s += w[k]; }
    const float inv = 1.f / s;
#pragma unroll
    for (int k = 0; k < 16; ++k) w[k] *= inv;

    // B-fragment element addresses: flat index ka*4 + (n&3), ka = kc*4+2*half.
    const int ncol = n & 3;

    v8f d = {};  // 16x16 f32 accumulator (8 VGPRs)
#pragma unroll
    for (int kc = 0; kc < 4; ++kc) {
        v2f a, b;
        // Constant-index selects (one v_cndmask each) instead of dynamic
        // register-array indexing.
        a.x = half ? w[kc * 4 + 2] : w[kc * 4 + 0];
        a.y = half ? w[kc * 4 + 3] : w[kc * 4 + 1];
        const int ka = kc * 4 + 2 * half;
        b.x = GATE_COEFFS_C[ka * 4 + ncol];
        b.y = GATE_COEFFS_C[(ka + 1) * 4 + ncol];
        d = __builtin_amdgcn_wmma_f32_16x16x4_f32(
            /*neg_a=*/false, a, /*neg_b=*/false, b,
            /*c_mod=*/(short)0, d, /*reuse_a=*/false, /*reuse_b=*/false);
    }

    // Write back D columns 0..3: coeffs[tile*16 + M][n], M = v + 8*half.
    if (n < 4) {
#pragma unroll
        for (int v = 0; v < 8; ++v) {
            const int M = v + 8 * half;
            coeffs[((size_t)(tile * 16 + M)) * 4 + n] = d[v];
        }
    }
}

// ---------------------------------------------------------------------------
// Phase 2: memory-bound gather + gate. One thread owns one output column j
// and ROWS batch rows: idx_a/idx_b/coeffs live in registers and are reused
// across ROWS rows (amortizes the 32 B/column metadata 16x). Gathers from x
// hit the 192 MB L2 (x is 128 MB); output stores are fully coalesced.
// 32 independent gather loads in flight per thread hide latency.
// ---------------------------------------------------------------------------
#define ROWS 16

__global__ __launch_bounds__(256) void gate_gather_kernel(
    const float* __restrict__ x,          // [BATCH][IN_DIM]
    const float* __restrict__ coeffs,     // [OUT_DIM][4]
    const long long* __restrict__ idx_a,  // [OUT_DIM]
    const long long* __restrict__ idx_b,  // [OUT_DIM]
    float* __restrict__ out,              // [BATCH][OUT_DIM]
    int batch, int in_dim, int out_dim)
{
    const int j = blockIdx.x * blockDim.x + threadIdx.x;
    if (j >= out_dim) return;
    const int i0 = blockIdx.y * ROWS;

    const float4 c = ((const float4*)coeffs)[j];   // c0, ca, cb, cab
    const int ia = (int)idx_a[j];
    const int ib = (int)idx_b[j];

    const float* __restrict__ xr = x + (size_t)i0 * in_dim;
    float* __restrict__ op = out + (size_t)i0 * out_dim + j;

    if (i0 + ROWS <= batch) {
        // Full tile: no per-row bounds checks.
#pragma unroll
        for (int r = 0; r < ROWS; ++r) {
            const float a = xr[ia];
            const float b = xr[ib];
            *op = fmaf(c.w, a * b, fmaf(c.z, b, fmaf(c.y, a, c.x)));
            xr += in_dim;
            op += out_dim;
        }
    } else {
        for (int r = 0; r < ROWS && (i0 + r) < batch; ++r) {
            const float a = xr[ia];
            const float b = xr[ib];
            *op = fmaf(c.w, a * b, fmaf(c.z, b, fmaf(c.y, a, c.x)));
            xr += in_dim;
            op += out_dim;
        }
    }
}

// ---------------------------------------------------------------------------
extern "C" void kernel_launch(void* const* d_in, const int* in_sizes, int n_in,
                              void* d_out, int out_size, void* d_ws, size_t ws_size,
                              hipStream_t stream)
{
    const float*     x       = (const float*)d_in[0];
    const float*     weights = (const float*)d_in[1];
    const long long* idx_a   = (const long long*)d_in[2];
    const long long* idx_b   = (const long long*)d_in[3];
    float*           out     = (float*)d_out;
    float*           coeffs  = (float*)d_ws;   // OUT_DIM*4 floats (128 KB)

    const int out_dim = in_sizes[1] / 16;       // weights is [OUT_DIM][16]
    const int batch   = out_size / out_dim;     // out is [BATCH][OUT_DIM]
    const int in_dim  = in_sizes[0] / batch;    // x is [BATCH][IN_DIM]

    // Phase 1: WMMA coeffs (one wave per 16 rows).
    coeffs_wmma_kernel<<<(out_dim + 15) / 16, 32, 0, stream>>>(weights, coeffs);

    // Phase 2: bandwidth-bound gather/gate.
    dim3 grid((out_dim + 255) / 256, (batch + ROWS - 1) / ROWS);
    gate_gather_kernel<<<grid, 256, 0, stream>>>(x, coeffs, idx_a, idx_b, out,
                                                 batch, in_dim, out_dim);
}